// HashEmbedder4D_40716289966788
// MI455X (gfx1250) — compile-verified
//
#include <hip/hip_runtime.h>
#include <math.h>

#define NLVL 16
#define TBL  (1u << 19)
#define TMASK (TBL - 1u)

struct LevelRes { float res[NLVL]; float rest[NLVL]; };

__global__ __launch_bounds__(256)
void hash4d_kernel(const float* __restrict__ xg,
                   const float* __restrict__ bminp,
                   const float* __restrict__ bmaxp,
                   const float* __restrict__ tables,
                   float* __restrict__ out,
                   int N, LevelRes lr)
{
    __shared__ float sres[2 * NLVL];
    const int t = threadIdx.x;
    if (t < NLVL) { sres[t] = lr.res[t]; sres[t + NLVL] = lr.rest[t]; }
    __syncthreads();

    const long long tid = (long long)blockIdx.x * blockDim.x + t;
    const long long total = (long long)N * NLVL;
    if (tid >= total) return;

    const int level = (int)(tid & (NLVL - 1));
    const long long p = tid >> 4;

    const float4 xv = ((const float4*)xg)[p];
    const float4 bn = *(const float4*)bminp;
    const float4 bx = *(const float4*)bmaxp;

    // Stream-ahead prefetch of x (gfx1250 global_prefetch_b8 path);
    // one prefetch per point (level-0 lane).
    if (level == 0 && (p + 4096) < (long long)N)
        __builtin_prefetch(xg + (p + 4096) * 4, 0, 0);

    const float xd[4] = {xv.x, xv.y, xv.z, xv.w};
    const float bm[4] = {bn.x, bn.y, bn.z, bn.w};
    const float bM[4] = {bx.x, bx.y, bx.z, bx.w};
    const float rs = sres[level], rt = sres[NLVL + level];
    const float rr[4] = {rs, rs, rs, rt};

    unsigned blu[4];
    float w[4];
    bool keep = true;
    #pragma unroll
    for (int d = 0; d < 4; ++d) {
        const float g    = (bM[d] - bm[d]) / rr[d];          // grid step
        const float xc   = fminf(fmaxf(xd[d], bm[d]), bM[d]); // clip(x)
        keep = keep && (xd[d] == xc);
        const float bl   = floorf((xc - bm[d]) / g);          // bottom-left idx
        const float vmin = bl * g + bm[d];                    // voxel min vertex
        w[d]   = (xd[d] - vmin) / g;                          // weight from UNCLAMPED x
        blu[d] = (unsigned)bl;
    }

    // Hash partials: prime*idx and prime*(idx+1) per dim (primes {1, ...}).
    const unsigned pa0 = blu[0];
    const unsigned pa1 = blu[0] + 1u;
    const unsigned pb0 = blu[1] * 2654435761u;
    const unsigned pb1 = (blu[1] + 1u) * 2654435761u;
    const unsigned pc0 = blu[2] * 805459861u;
    const unsigned pc1 = (blu[2] + 1u) * 805459861u;
    const unsigned pd0 = blu[3] * 3674653429u;
    const unsigned pd1 = (blu[3] + 1u) * 3674653429u;

    const float2* tab = (const float2*)tables + (size_t)level * TBL;

    // Issue all 16 gathers (global_load_b64) before interpolating.
    float2 e[16];
    #pragma unroll
    for (int c = 0; c < 16; ++c) {
        unsigned h = ((c & 8) ? pa1 : pa0) ^ ((c & 4) ? pb1 : pb0)
                   ^ ((c & 2) ? pc1 : pc0) ^ ((c & 1) ? pd1 : pd0);
        e[c] = tab[h & TMASK];
    }

    // Quadrilinear interpolation, reducing dim0 (corner bit 8) first,
    // exactly matching the reference's reshape(-1,2,2,2,2,F) loop order.
    #pragma unroll
    for (int c = 0; c < 8; ++c) {
        e[c].x = e[c].x * (1.0f - w[0]) + e[c + 8].x * w[0];
        e[c].y = e[c].y * (1.0f - w[0]) + e[c + 8].y * w[0];
    }
    #pragma unroll
    for (int c = 0; c < 4; ++c) {
        e[c].x = e[c].x * (1.0f - w[1]) + e[c + 4].x * w[1];
        e[c].y = e[c].y * (1.0f - w[1]) + e[c + 4].y * w[1];
    }
    #pragma unroll
    for (int c = 0; c < 2; ++c) {
        e[c].x = e[c].x * (1.0f - w[2]) + e[c + 2].x * w[2];
        e[c].y = e[c].y * (1.0f - w[2]) + e[c + 2].y * w[2];
    }
    float2 r;
    r.x = e[0].x * (1.0f - w[3]) + e[1].x * w[3];
    r.y = e[0].y * (1.0f - w[3]) + e[1].y * w[3];

    // Coalesced: 16 consecutive lanes cover one point's contiguous 128B row.
    *(float2*)(out + (size_t)p * 32 + (size_t)level * 2) = r;

    if (level == 0)
        out[(size_t)N * 32 + p] = keep ? 1.0f : 0.0f;
}

extern "C" void kernel_launch(void* const* d_in, const int* in_sizes, int n_in,
                              void* d_out, int out_size, void* d_ws, size_t ws_size,
                              hipStream_t stream) {
    const float* x      = (const float*)d_in[0];
    const float* bmin   = (const float*)d_in[1];
    const float* bmax   = (const float*)d_in[2];
    const float* tables = (const float*)d_in[3];
    const int N = in_sizes[0] / 4;

    // Mirror numpy: b = exp((log(512)-log(16))/15), res[i] = floor(16*b^i); same for time.
    LevelRes lr;
    const double b  = exp((log(512.0) - log(16.0)) / 15.0);
    const double bt = exp((log(32.0)  - log(2.0))  / 15.0);
    for (int i = 0; i < NLVL; ++i) {
        lr.res[i]  = (float)floor(16.0 * pow(b,  (double)i));
        lr.rest[i] = (float)floor(2.0  * pow(bt, (double)i));
    }

    const long long total  = (long long)N * NLVL;
    const int       block  = 256;
    const long long blocks = (total + block - 1) / block;
    hipLaunchKernelGGL(hash4d_kernel, dim3((unsigned)blocks), dim3(block), 0, stream,
                       x, bmin, bmax, tables, (float*)d_out, N, lr);
}